// WST_Loss_86706799771749
// MI455X (gfx1250) — compile-verified
//
#include <hip/hip_runtime.h>
#include <hip/hip_bf16.h>
#include <math.h>

// Problem constants (from reference): N=4, C=64, H=W=64, patch=9, hp=4.
// Centers: i,j in [4,60) -> 56x56. j tiled by 16: j0 in {4,20,36,52} (last tile half-valid).
#define CH    64
#define HW    64
#define CST   4096      // channel stride in floats (H*W)
#define NST   262144    // batch stride in floats (C*H*W)
#define NPIX  16384     // N*H*W pixels per tensor (4*64*64/... = 4*4096)

typedef __attribute__((ext_vector_type(2))) float v2f;
typedef __attribute__((ext_vector_type(8))) float v8f;

// ---------------------------------------------------------------------------
// Kernel 1: per-pixel inverse channel norms for warpedA and tensorB,
// and zero-init of the scalar output (d_out is poisoned by the harness).
// invnorm[p] = 1 / (||x[n,:,h,w]|| + 1e-8)
// ---------------------------------------------------------------------------
__global__ void wst_norms_kernel(const float* __restrict__ A,
                                 const float* __restrict__ B,
                                 float* __restrict__ invA,
                                 float* __restrict__ invB,
                                 float* __restrict__ out) {
    int t = blockIdx.x * blockDim.x + threadIdx.x;
    if (t == 0) out[0] = 0.0f;
    if (t >= 2 * NPIX) return;
    const float* src = (t < NPIX) ? A : B;
    float*       dst = (t < NPIX) ? invA : invB;
    int p  = t & (NPIX - 1);
    int n  = p >> 12;          // p / 4096
    int hw = p & 4095;
    const float* base = src + n * NST + hw;
    float s = 0.0f;
    #pragma unroll
    for (int c = 0; c < CH; ++c) {
        float v = base[c * CST];
        s += v * v;
    }
    dst[p] = 1.0f / (sqrtf(s) + 1e-8f);
}

// ---------------------------------------------------------------------------
// Kernel 2: fused local correlation (WMMA f32 16x16x4, K-stepped over C=64)
// + 9x9 patch gather + online log-sum-exp + loss accumulation.
// One wave (32 threads) per (n, center row i, 16-wide center j-tile).
// ---------------------------------------------------------------------------
__global__ __launch_bounds__(32)
void wst_corr_loss_kernel(const float* __restrict__ Aw,     // warpedA
                          const float* __restrict__ Bt,     // tensorB
                          const float* __restrict__ invA,
                          const float* __restrict__ invB,
                          float* __restrict__ out) {
    __shared__ float tile[16 * 32];   // D staging: 16 centers x 32 b-columns
    __shared__ float lred[16];

    const int lane = threadIdx.x;     // wave32
    const int half = lane >> 4;       // K-pair select for A/B fragments
    const int l16  = lane & 15;

    int b   = blockIdx.x;             // 4 * 56 * 4 blocks
    int n   = b / (56 * 4);
    int rem = b % (56 * 4);
    int i   = 4 + (rem >> 2);         // center row, in [4,60)
    int j0  = 4 + 16 * (rem & 3);     // first center column of this tile

    // ---- Preload A fragments for all 16 K-steps (ISA 16x4 f32 A layout):
    // lane = M (center index), half selects K pair {0,1} vs {2,3} within step.
    float ax[16], ay[16];
    {
        int jA = j0 + l16; if (jA > 63) jA = 63;       // clamp (unused centers)
        const float* ab = Aw + n * NST + i * HW + jA;
        #pragma unroll
        for (int s = 0; s < 16; ++s) {
            int kk = 4 * s + 2 * half;
            ax[s] = ab[kk * CST];
            ay[s] = ab[(kk + 1) * CST];
        }
    }

    const int  jc    = j0 + l16;                       // this lane's center col
    const bool valid = (lane < 16) && (jc < 60);
    const float ia   = valid ? invA[n * 4096 + i * 64 + jc] : 0.0f;

    float Mx = -INFINITY;   // running max   (online log-sum-exp)
    float se = 0.0f;        // running sum of exp
    float sx = 0.0f;        // running sum of raw values

    for (int di = -4; di <= 4; ++di) {
        int by = i + di;                               // b row, always in [0,64)
        v8f acc0 = {};                                 // b cols j0-4   .. j0+11
        v8f acc1 = {};                                 // b cols j0+12  .. j0+27
        const float* brow = Bt + n * NST + by * HW;
        int jb0 = j0 - 4 + l16;                        // >= 0, <= 63
        int jb1 = jb0 + 16; if (jb1 > 63) jb1 = 63;    // clamp (cols only used by invalid centers)
        const float* p0 = brow + jb0;
        const float* p1 = brow + jb1;

        #pragma unroll
        for (int s = 0; s < 16; ++s) {                 // K = 64 in steps of 4
            int kk = 4 * s + 2 * half;
            v2f af; af.x = ax[s];          af.y = ay[s];
            v2f b0; b0.x = p0[kk * CST];   b0.y = p0[(kk + 1) * CST];
            v2f b1; b1.x = p1[kk * CST];   b1.y = p1[(kk + 1) * CST];
            acc0 = __builtin_amdgcn_wmma_f32_16x16x4_f32(
                       false, af, false, b0, (short)0, acc0, false, false);
            acc1 = __builtin_amdgcn_wmma_f32_16x16x4_f32(
                       false, af, false, b1, (short)0, acc1, false, false);
        }

        // Stage D (16x32) to LDS: D VGPR r, lane -> (M = r + 8*half, N = l16)
        #pragma unroll
        for (int r = 0; r < 8; ++r) {
            tile[(r + 8 * half) * 32 + l16]      = acc0[r];
            tile[(r + 8 * half) * 32 + 16 + l16] = acc1[r];
        }
        __syncthreads();

        if (valid) {
            const float* ibrow = invB + n * 4096 + by * 64;
            #pragma unroll
            for (int dj = 0; dj < 9; ++dj) {
                int cb  = l16 + dj;                    // block col = (jc-4+dj) - (j0-4)
                float v = tile[l16 * 32 + cb] * ia * ibrow[jc - 4 + dj];
                float nm = fmaxf(Mx, v);
                se = se * __expf(Mx - nm) + __expf(v - nm);
                Mx = nm;
                sx += v;
            }
        }
        __syncthreads();
    }

    // loss contribution per valid center: sum_k (lse - x_k) = 81*lse - sum_k x_k
    float loss = valid ? (81.0f * (Mx + __logf(se)) - sx) : 0.0f;
    if (lane < 16) lred[lane] = loss;
    __syncthreads();
    if (lane == 0) {
        float t = 0.0f;
        #pragma unroll
        for (int k = 0; k < 16; ++k) t += lred[k];
        atomicAdd(out, t);
    }
}

// ---------------------------------------------------------------------------
// Launch: inputs are {featsA, warpedA, tensorB, warp_grid, patch_size}.
// Reference only consumes warpedA (d_in[1]) and tensorB (d_in[2]); patch=9.
// ---------------------------------------------------------------------------
extern "C" void kernel_launch(void* const* d_in, const int* in_sizes, int n_in,
                              void* d_out, int out_size, void* d_ws, size_t ws_size,
                              hipStream_t stream) {
    const float* warpedA = (const float*)d_in[1];
    const float* tensorB = (const float*)d_in[2];
    float* out  = (float*)d_out;
    float* invA = (float*)d_ws;           // 16384 floats
    float* invB = invA + NPIX;            // 16384 floats

    wst_norms_kernel<<<(2 * NPIX + 255) / 256, 256, 0, stream>>>(
        warpedA, tensorB, invA, invB, out);

    // 4 batches * 56 center rows * 4 j-tiles = 896 single-wave workgroups
    wst_corr_loss_kernel<<<4 * 56 * 4, 32, 0, stream>>>(
        warpedA, tensorB, invA, invB, out);
}